// TrainingImageLayer_59734405153334
// MI455X (gfx1250) — compile-verified
//
#include <hip/hip_runtime.h>
#include <math.h>
#include <stdint.h>

// ---------------- problem constants (match reference) ----------------
#define H_OUT   400
#define W_OUT   400
#define PSF_W   121
#define MARGIN  44
#define IM      488            // H + 2*MARGIN
#define NZ      64
#define N_EMIT  2048
#define Z_MIN   (-2.0f)
#define Z_RANGE 4.0f
#define SIGMA2  100.0f         // SIGMA_READ^2
#define MAG_OVER_PX (100.0f / 11.0f)

#define PSF_ELEMS (PSF_W * PSF_W)       // 14641
#define LSTRIDE   (PSF_W + 1)           // 122 (one zero pad row/col at index 0)
#define LDS_ELEMS (LSTRIDE * LSTRIDE)   // 14884 floats = 59536 B  (< 320KB/WGP, ~5 blocks/WGP)

#define THREADS 256
#define NWAVES  (THREADS / 32)

// ---------------- CDNA5 async global->LDS path (guarded) ----------------
// Builtin signature (from hipcc diagnostic): (AS1 int*, AS3 int*, imm offset, imm cpol)
typedef __attribute__((address_space(1))) int* as1i_t;
typedef __attribute__((address_space(3))) int* as3i_t;

#if defined(__has_builtin)
#  if __has_builtin(__builtin_amdgcn_global_load_async_to_lds_b32)
#    define HAVE_ASYNC_LDS 1
#  endif
#  if __has_builtin(__builtin_amdgcn_s_wait_asynccnt)
#    define HAVE_WAIT_ASYNC 1
#  endif
#endif

__device__ __forceinline__ void wait_async0() {
#if defined(HAVE_WAIT_ASYNC)
  __builtin_amdgcn_s_wait_asynccnt(0);
#else
  asm volatile("s_wait_asynccnt 0" ::: "memory");
#endif
}

// order-preserving float <-> uint keys for atomic min/max
__device__ __forceinline__ unsigned fkey(float f) {
  unsigned u = __float_as_uint(f);
  return (u & 0x80000000u) ? ~u : (u | 0x80000000u);
}
__device__ __forceinline__ float funkey(unsigned k) {
  return (k & 0x80000000u) ? __uint_as_float(k & 0x7fffffffu)
                           : __uint_as_float(~k);
}

// ---------------- kernel 1: zero canvas + init min/max ----------------
__global__ __launch_bounds__(256)
void k_init(float* __restrict__ canvas, unsigned* __restrict__ mm, int n) {
  int i = blockIdx.x * 256 + threadIdx.x;
  if (i < n) canvas[i] = 0.0f;
  if (i == 0) { mm[0] = 0xFFFFFFFFu; mm[1] = 0u; }
}

// ---------------- kernel 2: per-emitter shifted-PSF scatter ----------------
__global__ __launch_bounds__(THREADS)
void k_scatter(const float* __restrict__ xyz,     // (N_EMIT,3)
               const float* __restrict__ nphot,   // (N_EMIT)
               const float* __restrict__ psf_bank,// (NZ,121,121)
               float* __restrict__ canvas) {      // (IM,IM)
  __shared__ float   L[LDS_ELEMS];
  __shared__ float   red[NWAVES];
  __shared__ float   s_scale;

  const int e   = blockIdx.x;
  const int tid = threadIdx.x;

  const float x   = xyz[3 * e + 0];
  const float y   = xyz[3 * e + 1];
  const float z   = xyz[3 * e + 2];
  const float nph = nphot[e];

  // z -> PSF slice index (round-half-even like jnp.round, then clip)
  float zf = rintf((z - Z_MIN) * ((float)(NZ - 1) / Z_RANGE));
  zf = fminf(fmaxf(zf, 0.0f), (float)(NZ - 1));
  const int zi = (int)zf;

  const float lx  = x * MAG_OVER_PX;
  const float ly  = y * MAG_OVER_PX;
  const float off = (float)IM * 0.5f - (float)PSF_W * 0.5f;  // 183.5
  const float r0f = ly + off;
  const float c0f = lx + off;
  const float r0fl = floorf(r0f), c0fl = floorf(c0f);
  const int   r0 = (int)r0fl, c0 = (int)c0fl;
  const float rs = r0f - r0fl, cs = c0f - c0fl;

  const float w00 = (1.0f - rs) * (1.0f - cs);  // psf[i  ][j  ]
  const float w10 = rs * (1.0f - cs);           // psf[i-1][j  ]
  const float w01 = (1.0f - rs) * cs;           // psf[i  ][j-1]
  const float w11 = rs * cs;                    // psf[i-1][j-1]

  // zero pad row 0 and column 0
  for (int j = tid; j < LSTRIDE; j += THREADS) L[j] = 0.0f;
  for (int i = tid; i < LSTRIDE; i += THREADS) L[i * LSTRIDE] = 0.0f;

  const float* psf = psf_bank + (size_t)zi * PSF_ELEMS;

  // stage PSF into LDS (padded by 1 row/col) via CDNA5 async global->LDS copy
#if defined(HAVE_ASYNC_LDS)
  for (int idx = tid; idx < PSF_ELEMS; idx += THREADS) {
    const int i = idx / PSF_W;
    const int j = idx - i * PSF_W;
    __builtin_amdgcn_global_load_async_to_lds_b32(
        (as1i_t)(psf + idx),
        (as3i_t)&L[(i + 1) * LSTRIDE + (j + 1)],
        0, 0);
  }
  wait_async0();
#else
  for (int idx = tid; idx < PSF_ELEMS; idx += THREADS) {
    const int i = idx / PSF_W;
    const int j = idx - i * PSF_W;
    L[(i + 1) * LSTRIDE + (j + 1)] = psf[idx];
  }
#endif
  __syncthreads();

  // ---- pass 1: sum of |bilinear shift| for normalization ----
  float sum = 0.0f;
  for (int idx = tid; idx < PSF_ELEMS; idx += THREADS) {
    const int i = idx / PSF_W;
    const int j = idx - i * PSF_W;
    const float* p = &L[i * LSTRIDE + j];
    // p[0]=psf[i-1][j-1]  p[1]=psf[i-1][j]  p[LSTRIDE]=psf[i][j-1]  p[LSTRIDE+1]=psf[i][j]
    float v = w11 * p[0] + w10 * p[1] + w01 * p[LSTRIDE] + w00 * p[LSTRIDE + 1];
    sum += fabsf(v);
  }
  // wave32 reduce, then cross-wave via LDS
  #pragma unroll
  for (int m = 16; m >= 1; m >>= 1) sum += __shfl_xor(sum, m, 32);
  if ((tid & 31) == 0) red[tid >> 5] = sum;
  __syncthreads();
  if (tid == 0) {
    float t = 0.0f;
    #pragma unroll
    for (int w = 0; w < NWAVES; ++w) t += red[w];
    s_scale = nph / t;
  }
  __syncthreads();

  // fold normalization scale into the bilinear weights (scale > 0 so |v|*s == |v*s|)
  const float scale = s_scale;
  const float sw00 = w00 * scale;
  const float sw10 = w10 * scale;
  const float sw01 = w01 * scale;
  const float sw11 = w11 * scale;

  // ---- pass 2: scatter-add scaled tile into canvas (L2-resident atomics) ----
  for (int idx = tid; idx < PSF_ELEMS; idx += THREADS) {
    const int i = idx / PSF_W;
    const int j = idx - i * PSF_W;
    const float* p = &L[i * LSTRIDE + j];
    float v = sw11 * p[0] + sw10 * p[1] + sw01 * p[LSTRIDE] + sw00 * p[LSTRIDE + 1];
    const float val = fabsf(v);
    const int row = r0 + i;
    const int col = c0 + j;
    if ((unsigned)row < (unsigned)IM && (unsigned)col < (unsigned)IM)
      atomicAdd(&canvas[row * IM + col], val);
  }
}

// ---------------- kernel 3: crop + noise (into d_out) + global min/max ----------------
__global__ __launch_bounds__(256)
void k_noise(const float* __restrict__ canvas, const float* __restrict__ noise,
             float* __restrict__ out, unsigned* __restrict__ mm) {
  const int idx = blockIdx.x * 256 + threadIdx.x;
  unsigned kmin = 0xFFFFFFFFu, kmax = 0u;
  if (idx < H_OUT * W_OUT) {
    const int r = idx / W_OUT;
    const int c = idx - r * W_OUT;
    const float im = canvas[(r + MARGIN) * IM + (c + MARGIN)];
    const float v  = fmaf(sqrtf(fmaxf(im, 0.0f) + SIGMA2), noise[idx], im);
    out[idx] = v;
    const unsigned k = fkey(v);
    kmin = k; kmax = k;
  }
  #pragma unroll
  for (int m = 16; m >= 1; m >>= 1) {
    unsigned omin = __shfl_xor(kmin, m, 32);
    unsigned omax = __shfl_xor(kmax, m, 32);
    kmin = (omin < kmin) ? omin : kmin;
    kmax = (omax > kmax) ? omax : kmax;
  }
  __shared__ unsigned smin[8], smax[8];
  const int w = threadIdx.x >> 5;
  if ((threadIdx.x & 31) == 0) { smin[w] = kmin; smax[w] = kmax; }
  __syncthreads();
  if (threadIdx.x == 0) {
    #pragma unroll
    for (int i = 1; i < 8; ++i) {
      if (smin[i] < kmin) kmin = smin[i];
      if (smax[i] > kmax) kmax = smax[i];
    }
    atomicMin(&mm[0], kmin);
    atomicMax(&mm[1], kmax);
  }
}

// ---------------- kernel 4: min-max normalize d_out in place ----------------
__global__ __launch_bounds__(256)
void k_norm(float* __restrict__ out, const unsigned* __restrict__ mm, int n) {
  const int i = blockIdx.x * 256 + threadIdx.x;
  if (i < n) {
    const float mn  = funkey(mm[0]);
    const float mx  = funkey(mm[1]);
    const float inv = 1.0f / (mx - mn);
    out[i] = (out[i] - mn) * inv;
  }
}

// ---------------- host-side launch ----------------
extern "C" void kernel_launch(void* const* d_in, const int* in_sizes, int n_in,
                              void* d_out, int out_size, void* d_ws, size_t ws_size,
                              hipStream_t stream) {
  const float* xyz      = (const float*)d_in[0];   // (1, N_EMIT, 3)
  const float* nphot    = (const float*)d_in[1];   // (1, N_EMIT)
  // d_in[2] = xy_center: unused by the reference computation
  const float* psf_bank = (const float*)d_in[3];   // (NZ, 121, 121)
  const float* noise    = (const float*)d_in[4];   // (H, W)
  float* out = (float*)d_out;

  char* ws = (char*)d_ws;
  float*    canvas = (float*)ws;                                        // IM*IM f32
  unsigned* mm     = (unsigned*)(ws + sizeof(float) * (size_t)(IM * IM)); // 2 keys

  const int ncanvas = IM * IM;
  const int nout    = H_OUT * W_OUT;

  k_init<<<(ncanvas + 255) / 256, 256, 0, stream>>>(canvas, mm, ncanvas);
  k_scatter<<<N_EMIT, THREADS, 0, stream>>>(xyz, nphot, psf_bank, canvas);
  k_noise<<<(nout + 255) / 256, 256, 0, stream>>>(canvas, noise, out, mm);
  k_norm<<<(nout + 255) / 256, 256, 0, stream>>>(out, mm, nout);
}